// DrosophilaOpticLobeCircuit_51470888075855
// MI455X (gfx1250) — compile-verified
//
#include <hip/hip_runtime.h>
#include <stdint.h>

// ---------------- problem constants (match reference) ----------------
#define N_NEURONS 140000
#define N_TM1     20000
#define N_TYPES   7
#define N_EDGES   2240000
#define BATCH     8
#define DT_F      0.1f
#define STEPS     100

// ---------------- tiling ----------------
#define BS        256
#define NB_N      ((N_NEURONS + BS - 1) / BS)          // 547
#define NB_E      ((N_EDGES  + BS - 1) / BS)           // 8750
#define ROWS      (N_NEURONS - N_TM1)                  // 120000
#define NB_STEP   ((ROWS + BS - 1) / BS)               // 469
#define NB_OUT    ((N_NEURONS * BATCH + BS - 1) / BS)  // 4375
#define CAP_EDGES 5120                                 // 40KB LDS edge stage / block (even)

// ---------------- CDNA5 async global->LDS helpers ----------------
__device__ __forceinline__ void async_pair_to_lds(uint32_t lds_off, const void* g) {
    // GLOBAL_LOAD_ASYNC_TO_LDS_B128 (VGLOBAL op 98): VDST = LDS byte offset VGPR,
    // VADDR = 64-bit global address VGPR pair, GV mode. Tracked by ASYNCcnt.
    asm volatile("global_load_async_to_lds_b128 %0, %1, off"
                 :: "v"(lds_off), "v"((unsigned long long)(uintptr_t)g)
                 : "memory");
}
__device__ __forceinline__ void wait_asynccnt0() {
    asm volatile("s_wait_asynccnt 0x0" ::: "memory");
}

// ---------------- CSR construction ----------------
__global__ __launch_bounds__(BS) void k_zero_deg(int* __restrict__ deg) {
    int i = blockIdx.x * BS + threadIdx.x;
    if (i < N_NEURONS) deg[i] = 0;
}

__global__ __launch_bounds__(BS) void k_count(const int* __restrict__ tgt, int* __restrict__ deg) {
    int e = blockIdx.x * BS + threadIdx.x;
    if (e < N_EDGES) atomicAdd(&deg[tgt[e]], 1);
}

__global__ __launch_bounds__(BS) void k_scan1(const int* __restrict__ deg, int* __restrict__ partials) {
    __shared__ int sm[BS];
    int i = blockIdx.x * BS + threadIdx.x;
    sm[threadIdx.x] = (i < N_NEURONS) ? deg[i] : 0;
    __syncthreads();
    for (int off = BS / 2; off > 0; off >>= 1) {
        if (threadIdx.x < (unsigned)off) sm[threadIdx.x] += sm[threadIdx.x + off];
        __syncthreads();
    }
    if (threadIdx.x == 0) partials[blockIdx.x] = sm[0];
}

__global__ __launch_bounds__(1024) void k_scan2(const int* __restrict__ partials,
                                                int* __restrict__ expart,
                                                int* __restrict__ row_start) {
    __shared__ int sm[1024];
    const int i = threadIdx.x;
    const int v = (i < NB_N) ? partials[i] : 0;
    sm[i] = v;
    __syncthreads();
    for (int off = 1; off < 1024; off <<= 1) {
        int add = (i >= off) ? sm[i - off] : 0;
        __syncthreads();
        sm[i] += add;
        __syncthreads();
    }
    if (i < NB_N) expart[i] = sm[i] - v;                  // exclusive prefix of block sums
    if (i == NB_N - 1) row_start[N_NEURONS] = sm[i];      // total == N_EDGES
}

__global__ __launch_bounds__(BS) void k_scan3(const int* __restrict__ deg,
                                              const int* __restrict__ expart,
                                              int* __restrict__ row_start,
                                              int* __restrict__ cursor) {
    __shared__ int sm[BS];
    int i = blockIdx.x * BS + threadIdx.x;
    int v = (i < N_NEURONS) ? deg[i] : 0;
    sm[threadIdx.x] = v;
    __syncthreads();
    for (int off = 1; off < BS; off <<= 1) {
        int add = (threadIdx.x >= (unsigned)off) ? sm[threadIdx.x - off] : 0;
        __syncthreads();
        sm[threadIdx.x] += add;
        __syncthreads();
    }
    if (i < N_NEURONS) {
        int rs = expart[blockIdx.x] + sm[threadIdx.x] - v;  // global exclusive prefix
        row_start[i] = rs;
        cursor[i]    = rs;
    }
}

// Pre-scale weights by sign-dependent scales and pack edges {src, w} by target.
__global__ __launch_bounds__(BS) void k_fill(const float* __restrict__ w,
                                             const int* __restrict__ src,
                                             const int* __restrict__ tgt,
                                             const float* __restrict__ se_p,
                                             const float* __restrict__ si_p,
                                             int* __restrict__ cursor,
                                             int2* __restrict__ csr) {
    int e = blockIdx.x * BS + threadIdx.x;
    if (e >= N_EDGES) return;
    const float se = fmaxf(se_p[0], 0.0f);
    const float si = fmaxf(si_p[0], 0.0f);
    const float wv = w[e];
    const float sw = (wv > 0.0f) ? wv * se : ((wv < 0.0f) ? wv * si : 0.0f);
    const int pos = atomicAdd(&cursor[tgt[e]], 1);
    csr[pos] = make_int2(src[e], __float_as_int(sw));
}

// ---------------- state init: v layout is [neuron][batch] (32B/neuron) ----------------
__global__ __launch_bounds__(BS) void k_init(const float* __restrict__ tm1,
                                             float* __restrict__ v,
                                             float* __restrict__ r0,
                                             float* __restrict__ r1) {
    int n = blockIdx.x * BS + threadIdx.x;
    if (n >= N_NEURONS) return;
    float vals[BATCH];
    if (n < N_TM1) {
        #pragma unroll
        for (int b = 0; b < BATCH; ++b) vals[b] = tm1[b * N_TM1 + n];
    } else {
        #pragma unroll
        for (int b = 0; b < BATCH; ++b) vals[b] = 0.0f;
    }
    #pragma unroll
    for (int b = 0; b < BATCH; ++b) {
        const size_t o = (size_t)n * BATCH + b;
        const float rr = fmaxf(vals[b], 0.0f);
        v[o] = vals[b];
        r0[o] = rr;
        r1[o] = rr;
    }
}

// 8-wide FMA of one edge into the batch accumulators
__device__ __forceinline__ void edge_fma(const int2 ed, const float* __restrict__ r_cur,
                                         float4& a0, float4& a1) {
    const float wv = __int_as_float(ed.y);
    const float4* rp = (const float4*)(r_cur + (size_t)ed.x * BATCH);
    const float4 q0 = rp[0];
    const float4 q1 = rp[1];
    a0.x = fmaf(wv, q0.x, a0.x);
    a0.y = fmaf(wv, q0.y, a0.y);
    a0.z = fmaf(wv, q0.z, a0.z);
    a0.w = fmaf(wv, q0.w, a0.w);
    a1.x = fmaf(wv, q1.x, a1.x);
    a1.y = fmaf(wv, q1.y, a1.y);
    a1.z = fmaf(wv, q1.z, a1.z);
    a1.w = fmaf(wv, q1.w, a1.w);
}

// ---------------- fused step: gather (CSR, LDS-staged) + integrate + relu ----------------
__global__ __launch_bounds__(BS) void k_step(float* __restrict__ v,
                                             const float* __restrict__ r_cur,
                                             float* __restrict__ r_nxt,
                                             const int2* __restrict__ csr,
                                             const int* __restrict__ row_start,
                                             const float* __restrict__ bias,
                                             const float* __restrict__ tau) {
    __shared__ int2  s_edges[CAP_EDGES];
    __shared__ float s_dtinv[N_TYPES];
    __shared__ int   s_seg[2];

    const int tid   = threadIdx.x;
    const int first = N_TM1 + blockIdx.x * BS;    // rows [first, first+BS) — all non-Tm1
    const int n     = first + tid;

    if (tid < N_TYPES) {
        float t = tau[tid];
        t = fminf(fmaxf(t, 1.0f), 100.0f);
        s_dtinv[tid] = DT_F / t;
    }
    if (tid == 0) s_seg[0] = row_start[first];
    if (tid == 1) {
        int last = first + BS;
        if (last > N_NEURONS) last = N_NEURONS;
        s_seg[1] = row_start[last];
    }
    __syncthreads();

    // Stage this block's contiguous CSR segment into LDS with 16B async copies.
    // Round the start down to an even edge index so global/LDS addresses are 16B aligned.
    const int astart = s_seg[0] & ~1;
    int count = s_seg[1] - astart;
    if (count > CAP_EDGES) count = CAP_EDGES;     // overflow rows fall back to global reads
    const int npairs = (count + 1) >> 1;          // may over-copy 1 edge (csr is padded)

    const uint32_t lds_base = (uint32_t)(uintptr_t)(&s_edges[0]); // low 32b == LDS offset
    const char* gbase = (const char*)(csr + astart);
    for (int i = tid; i < npairs; i += BS) {
        async_pair_to_lds(lds_base + (uint32_t)i * 16u, (const void*)(gbase + (size_t)i * 16));
    }
    wait_asynccnt0();
    __syncthreads();

    if (n < N_NEURONS) {
        const int s = row_start[n];
        const int e = row_start[n + 1];
        float4 a0 = make_float4(0.f, 0.f, 0.f, 0.f);
        float4 a1 = make_float4(0.f, 0.f, 0.f, 0.f);
        if (e - astart <= count) {
            // fast path: this row's edges are fully staged in LDS
            const int2* ep = &s_edges[s - astart];
            for (int i = s; i < e; ++i, ++ep) {
                edge_fma(*ep, r_cur, a0, a1);
            }
        } else {
            // rare overflow path: read straight from L2-resident CSR
            for (int i = s; i < e; ++i) {
                edge_fma(csr[i], r_cur, a0, a1);
            }
        }
        float4* vp = (float4*)(v + (size_t)n * BATCH);
        float4 v0 = vp[0];
        float4 v1 = vp[1];
        const float bn = bias[n];
        const float k  = s_dtinv[n / N_TM1];
        // v += dt/tau * (-v + syn + bias)
        v0.x += k * (bn + a0.x - v0.x);
        v0.y += k * (bn + a0.y - v0.y);
        v0.z += k * (bn + a0.z - v0.z);
        v0.w += k * (bn + a0.w - v0.w);
        v1.x += k * (bn + a1.x - v1.x);
        v1.y += k * (bn + a1.y - v1.y);
        v1.z += k * (bn + a1.z - v1.z);
        v1.w += k * (bn + a1.w - v1.w);
        vp[0] = v0;
        vp[1] = v1;
        float4* rp = (float4*)(r_nxt + (size_t)n * BATCH);
        rp[0] = make_float4(fmaxf(v0.x, 0.f), fmaxf(v0.y, 0.f), fmaxf(v0.z, 0.f), fmaxf(v0.w, 0.f));
        rp[1] = make_float4(fmaxf(v1.x, 0.f), fmaxf(v1.y, 0.f), fmaxf(v1.z, 0.f), fmaxf(v1.w, 0.f));
    }
}

// ---------------- transpose [neuron][batch] -> reference (batch, neuron) ----------------
__global__ __launch_bounds__(BS) void k_out(const float* __restrict__ v, float* __restrict__ out) {
    int idx = blockIdx.x * BS + threadIdx.x;
    if (idx >= N_NEURONS * BATCH) return;
    const int n = idx >> 3;
    const int b = idx & 7;
    out[(size_t)b * N_NEURONS + n] = v[idx];
}

// ---------------- host launcher ----------------
extern "C" void kernel_launch(void* const* d_in, const int* in_sizes, int n_in,
                              void* d_out, int out_size, void* d_ws, size_t ws_size,
                              hipStream_t stream) {
    (void)in_sizes; (void)n_in; (void)out_size; (void)ws_size;
    const float* tm1  = (const float*)d_in[0];
    const float* wts  = (const float*)d_in[1];
    const float* bias = (const float*)d_in[2];
    const float* tau  = (const float*)d_in[3];
    const float* se_p = (const float*)d_in[4];
    const float* si_p = (const float*)d_in[5];
    const int*   src  = (const int*)d_in[6];
    const int*   tgt  = (const int*)d_in[7];
    float* out = (float*)d_out;

    // workspace carve (256B aligned): ~33 MB total
    char* p = (char*)d_ws;
    auto carve = [&](size_t bytes) -> char* {
        char* q = p;
        p += (bytes + 255) & ~(size_t)255;
        return q;
    };
    float* v         = (float*)carve((size_t)N_NEURONS * BATCH * 4);
    float* r0        = (float*)carve((size_t)N_NEURONS * BATCH * 4);
    float* r1        = (float*)carve((size_t)N_NEURONS * BATCH * 4);
    int*   deg       = (int*)  carve((size_t)N_NEURONS * 4);
    int*   row_start = (int*)  carve((size_t)(N_NEURONS + 1) * 4);
    int*   cursor    = (int*)  carve((size_t)N_NEURONS * 4);
    int*   partials  = (int*)  carve((size_t)NB_N * 4);
    int*   expart    = (int*)  carve((size_t)NB_N * 4);
    int2*  csr       = (int2*) carve((size_t)(N_EDGES + 2) * 8);  // +2 pad for b128 stage

    // ---- build CSR (by target) once per launch ----
    k_zero_deg<<<NB_N, BS, 0, stream>>>(deg);
    k_count<<<NB_E, BS, 0, stream>>>(tgt, deg);
    k_scan1<<<NB_N, BS, 0, stream>>>(deg, partials);
    k_scan2<<<1, 1024, 0, stream>>>(partials, expart, row_start);
    k_scan3<<<NB_N, BS, 0, stream>>>(deg, expart, row_start, cursor);
    k_fill<<<NB_E, BS, 0, stream>>>(wts, src, tgt, se_p, si_p, cursor, csr);

    // ---- init state ----
    k_init<<<NB_N, BS, 0, stream>>>(tm1, v, r0, r1);

    // ---- 100 fused steps, double-buffered rates ----
    float* rc = r0;
    float* rn = r1;
    for (int t = 0; t < STEPS; ++t) {
        k_step<<<NB_STEP, BS, 0, stream>>>(v, rc, rn, csr, row_start, bias, tau);
        float* tmp = rc; rc = rn; rn = tmp;
    }

    // ---- emit (BATCH, N_NEURONS) ----
    k_out<<<NB_OUT, BS, 0, stream>>>(v, out);
}